// OpticFlowMask_81295140979376
// MI455X (gfx1250) — compile-verified
//
#include <hip/hip_runtime.h>

// Problem constants (match reference)
#define BB   12
#define HH   192
#define WW   640
#define HW   (HH * WW)        // 122880  (divisible by 256)
#define BHW  (BB * HW)        // 1474560
#define EPSF 1e-7f
#define THRESH 0.98f
#define TPB  256              // 8 wave32 waves per block

typedef __attribute__((ext_vector_type(2))) float v2f;
typedef __attribute__((ext_vector_type(8))) float v8f;

// ---------- monotone float<->uint encoding so uint atomicMin/Max == float min/max ----------
__device__ __forceinline__ unsigned enc_f32(float f) {
  unsigned u = __float_as_uint(f);
  return (u & 0x80000000u) ? ~u : (u | 0x80000000u);
}
__device__ __forceinline__ float dec_f32(unsigned u) {
  return __uint_as_float((u & 0x80000000u) ? (u ^ 0x80000000u) : ~u);
}

// ---------- V_WMMA_F32_16X16X4_F32 wrapper (builtin if declared, else inline asm) ----------
__device__ __forceinline__ v8f wmma_f32_16x16x4(v2f A, v2f B, v8f C) {
#if __has_builtin(__builtin_amdgcn_wmma_f32_16x16x4_f32)
  return __builtin_amdgcn_wmma_f32_16x16x4_f32(false, A, false, B, (short)0, C, false, false);
#else
  v8f D;
  asm volatile("v_wmma_f32_16x16x4_f32 %0, %1, %2, %3"
               : "=v"(D)
               : "v"(A), "v"(B), "v"(C));
  return D;
#endif
}

// ---------- per-batch projection coefficients: cp = depth*(Q @ [x,y,1]) + T ----------
// Q = (K@pose)[:3,:3] @ invK[:3,:3],  T = (K@pose)[:3,3]
__device__ __forceinline__ void calc_QT(const float* __restrict__ Km,
                                        const float* __restrict__ invKm,
                                        const float* __restrict__ posem,
                                        float Q[9], float T[3]) {
  float M[12];
#pragma unroll
  for (int i = 0; i < 3; ++i)
#pragma unroll
    for (int j = 0; j < 4; ++j) {
      float s = 0.f;
#pragma unroll
      for (int k = 0; k < 4; ++k) s = fmaf(Km[i * 4 + k], posem[k * 4 + j], s);
      M[i * 4 + j] = s;
    }
#pragma unroll
  for (int i = 0; i < 3; ++i) {
#pragma unroll
    for (int j = 0; j < 3; ++j) {
      float s = 0.f;
#pragma unroll
      for (int k = 0; k < 3; ++k) s = fmaf(M[i * 4 + k], invKm[k * 4 + j], s);
      Q[i * 3 + j] = s;
    }
    T[i] = M[i * 4 + 3];
  }
}

// ---------- projection via WMMA: two 16-pixel tiles per wave, K=4 GEMM ----------
// A (16x4 f32): lanes 0-15 row M=r hold {K0,K1}; lanes 16-31 hold {K2,K3}.
// B (4x16 f32): VGPR0 = K0 (lanes 0-15) / K2 (lanes 16-31); VGPR1 = K1 / K3.
// D (16x16 f32): lanes 0-15, VGPR j = row M=j for column N=lane.
__device__ __forceinline__ void project_wmma(const float Q[9], const float T[3],
                                             float px, float py, float d,
                                             float& cx, float& cy, float& cz) {
  const int lane = (int)(threadIdx.x & 31u);
  const int r = lane & 15;
  const bool lo = lane < 16;
  const float dx = d * px;
  const float dy = d * py;

  v2f A;
  A.x = lo ? (r == 0 ? Q[0] : r == 1 ? Q[3] : r == 2 ? Q[6] : 0.f)
           : (r == 0 ? Q[2] : r == 1 ? Q[5] : r == 2 ? Q[8] : 0.f);
  A.y = lo ? (r == 0 ? Q[1] : r == 1 ? Q[4] : r == 2 ? Q[7] : 0.f)
           : (r == 0 ? T[0] : r == 1 ? T[1] : r == 2 ? T[2] : 0.f);

  const float d_lo  = __shfl(d,  lane & 15, 32);  // depth of pixel (lane-16) for hi lanes
  const float dx_hi = __shfl(dx, lane | 16, 32);  // d*x of pixel (lane+16) for lo lanes
  const float dy_hi = __shfl(dy, lane | 16, 32);

  v2f B1;  // columns = pixels of lanes 0..15
  B1.x = lo ? dx : d_lo;
  B1.y = lo ? dy : 1.0f;
  v2f B2;  // columns = pixels of lanes 16..31
  B2.x = lo ? dx_hi : d;
  B2.y = lo ? dy_hi : 1.0f;

  v8f C = {0.f, 0.f, 0.f, 0.f, 0.f, 0.f, 0.f, 0.f};
  v8f D1 = wmma_f32_16x16x4(A, B1, C);
  v8f D2 = wmma_f32_16x16x4(A, B2, C);

  const float x2 = __shfl(D2[0], lane & 15, 32);
  const float y2 = __shfl(D2[1], lane & 15, 32);
  const float z2 = __shfl(D2[2], lane & 15, 32);
  cx = lo ? D1[0] : x2;
  cy = lo ? D1[1] : y2;
  cz = lo ? D1[2] : z2;
}

__device__ __forceinline__ void static_flow(int b, int n,
    const float* __restrict__ K, const float* __restrict__ invK,
    const float* __restrict__ pose, const float* __restrict__ depth,
    float& sx, float& sy) {
  float Q[9], T[3];
  calc_QT(K + b * 16, invK + b * 16, pose + b * 16, Q, T);
  const float d  = depth[(size_t)b * HW + n];
  const float px = (float)(n % WW);
  const float py = (float)(n / WW);
  float cx, cy, cz;
  project_wmma(Q, T, px, py, d, cx, cy, cz);
  const float iz = 1.0f / (cz + EPSF);
  sx = cx * iz - px;
  sy = cy * iz - py;
}

// ---------- block-wide min/max -> one global atomic per block ----------
__device__ __forceinline__ void block_minmax(float vmn, float vmx,
                                             unsigned* slot_mn, unsigned* slot_mx) {
#pragma unroll
  for (int off = 16; off > 0; off >>= 1) {
    vmn = fminf(vmn, __shfl_xor(vmn, off, 32));
    vmx = fmaxf(vmx, __shfl_xor(vmx, off, 32));
  }
  __shared__ float smn[TPB / 32], smx[TPB / 32];
  const int w = (int)(threadIdx.x >> 5);
  if ((threadIdx.x & 31u) == 0u) { smn[w] = vmn; smx[w] = vmx; }
  __syncthreads();
  if (threadIdx.x == 0) {
    float m = smn[0], M = smx[0];
#pragma unroll
    for (int i = 1; i < TPB / 32; ++i) { m = fminf(m, smn[i]); M = fmaxf(M, smx[i]); }
    atomicMin(slot_mn, enc_f32(m));
    atomicMax(slot_mx, enc_f32(M));
  }
  __syncthreads();  // allow reuse of smn/smx by a subsequent call
}

// ---------- shared check() computation (passes 2 & 3) ----------
__device__ __forceinline__ float compute_check(int b, int n,
    const float* __restrict__ flow, const float* __restrict__ depth,
    const float* __restrict__ K, const float* __restrict__ invK,
    const float* __restrict__ pose,
    float fmn, float fmx, float smn, float smx) {
  const float fx = flow[(size_t)b * 2 * HW + n];
  const float fy = flow[(size_t)b * 2 * HW + HW + n];
  float sx, sy;
  static_flow(b, n, K, invK, pose, depth, sx, sy);
  const float fs = 2.0f / (fmx - fmn);
  const float ss = 2.0f / (smx - smn);
  // the two "-1" offsets of the normalizations cancel in the difference
  const float dx = (fx - fmn) * fs - (sx - smn) * ss;
  const float dy = (fy - fmn) * fs - (sy - smn) * ss;
  return sqrtf(dx * dx + dy * dy);
}

// ================================ kernels ================================

// ws[0/1]=flow min/max, ws[2/3]=static min/max, ws[4/5]=check min/max (encoded)
__global__ void ofm_init(unsigned* __restrict__ ws) {
  const int t = (int)threadIdx.x;
  if (t < 3) { ws[2 * t] = 0xFFFFFFFFu; ws[2 * t + 1] = 0u; }
}

__global__ __launch_bounds__(TPB) void ofm_pass1(
    const float* __restrict__ flow, const float* __restrict__ depth,
    const float* __restrict__ K, const float* __restrict__ invK,
    const float* __restrict__ pose,
    float* __restrict__ out, unsigned* __restrict__ ws) {
  const int b = (int)blockIdx.y;
  const int n = (int)blockIdx.x * TPB + (int)threadIdx.x;
  const float fx = flow[(size_t)b * 2 * HW + n];
  const float fy = flow[(size_t)b * 2 * HW + HW + n];
  float sx, sy;
  static_flow(b, n, K, invK, pose, depth, sx, sy);
  // zero the scatter-target output regions (bwd_flow + seg_ref)
  out[(size_t)BHW + (size_t)(b * 2) * HW + n]     = 0.f;
  out[(size_t)BHW + (size_t)(b * 2 + 1) * HW + n] = 0.f;
  out[(size_t)3 * BHW + (size_t)b * HW + n]       = 0.f;
  block_minmax(fminf(fx, fy), fmaxf(fx, fy), ws + 0, ws + 1);
  block_minmax(fminf(sx, sy), fmaxf(sx, sy), ws + 2, ws + 3);
}

__global__ __launch_bounds__(TPB) void ofm_pass2(
    const float* __restrict__ flow, const float* __restrict__ depth,
    const float* __restrict__ K, const float* __restrict__ invK,
    const float* __restrict__ pose, unsigned* __restrict__ ws) {
  const int b = (int)blockIdx.y;
  const int n = (int)blockIdx.x * TPB + (int)threadIdx.x;
  const float fmn = dec_f32(ws[0]), fmx = dec_f32(ws[1]);
  const float smn = dec_f32(ws[2]), smx = dec_f32(ws[3]);
  const float chk = compute_check(b, n, flow, depth, K, invK, pose, fmn, fmx, smn, smx);
  block_minmax(chk, chk, ws + 4, ws + 5);
}

__global__ __launch_bounds__(TPB) void ofm_pass3(
    const float* __restrict__ flow, const float* __restrict__ depth,
    const float* __restrict__ K, const float* __restrict__ invK,
    const float* __restrict__ pose,
    float* __restrict__ out, const unsigned* __restrict__ ws) {
  const int b = (int)blockIdx.y;
  const int n = (int)blockIdx.x * TPB + (int)threadIdx.x;
  const float fmn = dec_f32(ws[0]), fmx = dec_f32(ws[1]);
  const float smn = dec_f32(ws[2]), smx = dec_f32(ws[3]);
  const float cmn = dec_f32(ws[4]), cmx = dec_f32(ws[5]);
  const float chk = compute_check(b, n, flow, depth, K, invK, pose, fmn, fmx, smn, smx);
  const float val = (chk - cmn) / (cmx - cmn);
  out[(size_t)b * HW + n] = (val < THRESH) ? 1.0f : 0.0f;
}

__global__ __launch_bounds__(TPB) void ofm_scatter(
    const float* __restrict__ flow, const int* __restrict__ seg,
    float* __restrict__ out) {
  const int b = (int)blockIdx.y;
  const int n = (int)blockIdx.x * TPB + (int)threadIdx.x;
  const float fx = flow[(size_t)b * 2 * HW + n];
  const float fy = flow[(size_t)b * 2 * HW + HW + n];
  const float px = (float)(n % WW);
  const float py = (float)(n / WW);
  int cxi = (int)rintf(px + fx);               // jnp.round == round-half-even
  int cyi = (int)rintf(py + fy);
  cxi = min(max(cxi, 0), WW - 1);
  cyi = min(max(cyi, 0), HH - 1);
  const int idx = cyi * WW + cxi;
  // last-write-wins scatter (duplicate-index order unspecified in reference)
  out[(size_t)BHW + (size_t)(b * 2) * HW + idx]     = -fx;
  out[(size_t)BHW + (size_t)(b * 2 + 1) * HW + idx] = -fy;
  out[(size_t)3 * BHW + (size_t)b * HW + idx]       = seg[(size_t)b * HW + n] ? 1.0f : 0.0f;
}

// ================================ launch ================================

extern "C" void kernel_launch(void* const* d_in, const int* in_sizes, int n_in,
                              void* d_out, int out_size, void* d_ws, size_t ws_size,
                              hipStream_t stream) {
  (void)in_sizes; (void)n_in; (void)out_size; (void)ws_size;
  const float* flow  = (const float*)d_in[0];  // [B,2,H,W]
  const float* depth = (const float*)d_in[1];  // [B,1,H,W]
  const float* K     = (const float*)d_in[2];  // [B,4,4]
  const float* invK  = (const float*)d_in[3];  // [B,4,4]
  const float* pose  = (const float*)d_in[4];  // [B,4,4]
  const int*   seg   = (const int*)d_in[5];    // [B,1,H,W]
  float*    out = (float*)d_out;               // [mask | bwd_flow | seg_ref]
  unsigned* ws  = (unsigned*)d_ws;             // 6 encoded min/max slots

  const dim3 grid(HW / TPB, BB);
  ofm_init<<<dim3(1), dim3(32), 0, stream>>>(ws);
  ofm_pass1<<<grid, TPB, 0, stream>>>(flow, depth, K, invK, pose, out, ws);
  ofm_pass2<<<grid, TPB, 0, stream>>>(flow, depth, K, invK, pose, ws);
  ofm_pass3<<<grid, TPB, 0, stream>>>(flow, depth, K, invK, pose, out, ws);
  ofm_scatter<<<grid, TPB, 0, stream>>>(flow, seg, out);
}